// BinaryDiff_40389872452190
// MI455X (gfx1250) — compile-verified
//
#include <hip/hip_runtime.h>

// BinaryDiff fused GEMM for MI455X (gfx1250, wave32, WMMA), round 4.
//
// out[M,N] = x[M,K] @ (base_T[K,N] + coeff * (2*mask[K,N] - 1))
// M = B*S = 8192, K = DIN = 4096, N = DOUT = 4096, all f32.
//
// Memory-bound (~402 MB HBM min vs 274 GFLOP) -> full-precision
// V_WMMA_F32_16X16X4_F32, sign-correction fused into the B-tile load.
// 256x256 block tiles keep L2-side traffic at ~4.3 GB. Double-buffered
// LDS software pipeline; A tile copied via CDNA5 async global->LDS
// (ASYNCcnt), B tile staged through VGPRs for the sign fusion.

typedef float v2f __attribute__((ext_vector_type(2)));
typedef float v8f __attribute__((ext_vector_type(8)));
typedef int   v4i __attribute__((ext_vector_type(4)));

#define BM 256
#define BN 256
#define BK 16
#define N_TOTAL 4096
#define K_TOTAL 4096
#define NTILES (K_TOTAL / BK)
#define NTHREADS 512

__device__ __forceinline__ void async_copy_b128(const float* gsrc, float* ldst) {
#if defined(__has_builtin) && __has_builtin(__builtin_amdgcn_global_load_async_to_lds_b128)
  // builtin signature: (v4i addrspace(1)*, v4i addrspace(3)*, imm off, imm cpol)
  typedef __attribute__((address_space(1))) v4i* gp_t;
  typedef __attribute__((address_space(3))) v4i* lp_t;
  gp_t g = (gp_t)(unsigned long long)gsrc;
  // addrspace(3) pointer value == 32-bit LDS byte offset (= addr[31:0])
  lp_t l = (lp_t)(unsigned)(unsigned long long)ldst;
  __builtin_amdgcn_global_load_async_to_lds_b128(g, l, 0, 0);
#else
  asm volatile("global_load_async_to_lds_b128 %0, %1, off"
               :
               : "v"((unsigned)(unsigned long long)ldst), "v"(gsrc)
               : "memory");
#endif
}

__device__ __forceinline__ void wait_async0() {
#if defined(__has_builtin) && __has_builtin(__builtin_amdgcn_s_wait_asynccnt)
  __builtin_amdgcn_s_wait_asynccnt(0);
#else
  asm volatile("s_wait_asynccnt 0x0" ::: "memory");
#endif
}

__global__ __launch_bounds__(NTHREADS) void binarydiff_wmma_gemm(
    const float* __restrict__ X,      // [M, K] row-major
    const float* __restrict__ baseT,  // [K, N] row-major
    const int*   __restrict__ mask,   // [K, N] row-major, {0,1}
    const float* __restrict__ coeff,  // [1]
    float* __restrict__ Out,          // [M, N] row-major
    int M)
{
  __shared__ float As[2][BM * BK];   // 2 x 16 KB
  __shared__ float Bs[2][BK * BN];   // 2 x 16 KB (fused W)

  const int tid   = threadIdx.x;
  const int lane  = tid & 31;
  const int wave  = tid >> 5;      // 0..15
  const int waveM = wave & 3;      // 0..3 -> 64-row strip
  const int waveN = wave >> 2;     // 0..3 -> 64-col strip
  const int lh    = lane >> 4;     // lane half: selects K pair (ISA 7.12.2)
  const int ll    = lane & 15;     // row (A) / col (B) within 16

  const int m0 = blockIdx.y * BM;
  const int n0 = blockIdx.x * BN;

  const float c = coeff[0];        // uniform scalar

  // ---- loader mappings (per thread) ----
  // A tile: 256x16 f32 = 1024 float4 -> 2 per thread
  // B tile:  16x256 f32 = 1024 float4 -> 2 per thread (+ int4 mask)
  int a_row[2], a_col[2], b_row[2], b_col[2];
#pragma unroll
  for (int r = 0; r < 2; ++r) {
    int idx  = tid + r * NTHREADS;
    a_row[r] = idx >> 2;            // BK/4 = 4 float4 per A row
    a_col[r] = (idx & 3) << 2;
    b_row[r] = idx >> 6;            // BN/4 = 64 float4 per B row
    b_col[r] = (idx & 63) << 2;
  }

  float4 bstage[2];
  int4   mstage[2];

  auto loadA_async = [&](int k0, int buf) {
#pragma unroll
    for (int r = 0; r < 2; ++r)
      async_copy_b128(X + (size_t)(m0 + a_row[r]) * K_TOTAL + (k0 + a_col[r]),
                      &As[buf][a_row[r] * BK + a_col[r]]);
  };
  auto loadB_global = [&](int k0) {
#pragma unroll
    for (int r = 0; r < 2; ++r) {
      size_t g  = (size_t)(k0 + b_row[r]) * N_TOTAL + (n0 + b_col[r]);
      bstage[r] = *(const float4*)(baseT + g);
      mstage[r] = *(const int4*)(mask + g);
    }
  };
  auto storeB_lds = [&](int buf) {
#pragma unroll
    for (int r = 0; r < 2; ++r) {
      float4 w;
      w.x = bstage[r].x + (mstage[r].x ? c : -c);
      w.y = bstage[r].y + (mstage[r].y ? c : -c);
      w.z = bstage[r].z + (mstage[r].z ? c : -c);
      w.w = bstage[r].w + (mstage[r].w ? c : -c);
      *(float4*)(&Bs[buf][b_row[r] * BN + b_col[r]]) = w;
    }
  };

  v8f acc[4][4];
  const v8f vzero = {};
#pragma unroll
  for (int mi = 0; mi < 4; ++mi)
#pragma unroll
    for (int ni = 0; ni < 4; ++ni)
      acc[mi][ni] = vzero;

  // ---- prologue: fill buffer 0 ----
  loadA_async(0, 0);
  loadB_global(0);
  storeB_lds(0);
  wait_async0();
  __syncthreads();

  for (int t = 0; t < NTILES; ++t) {
    const int buf = t & 1;
    const int k0  = t * BK;

    // issue next tile's loads (uniform branch: EXEC stays all-ones)
    if (t + 1 < NTILES) {
      loadA_async(k0 + BK, buf ^ 1);
      loadB_global(k0 + BK);
    }

    // ---- 4 WMMA k-steps of K=4 each; 16 WMMAs per step ----
#pragma unroll
    for (int ks = 0; ks < BK / 4; ++ks) {
      const int kk = 4 * ks + 2 * lh;     // this lane-half's K pair

      v2f afrag[4];
#pragma unroll
      for (int mi = 0; mi < 4; ++mi)
        afrag[mi] =
            *(const v2f*)(&As[buf][(waveM * 64 + mi * 16 + ll) * BK + kk]);

      v2f bfrag[4];
#pragma unroll
      for (int ni = 0; ni < 4; ++ni) {
        const float* bp = &Bs[buf][kk * BN + (waveN * 64 + ni * 16 + ll)];
        v2f tv;
        tv.x = bp[0];
        tv.y = bp[BN];
        bfrag[ni] = tv;
      }

#pragma unroll
      for (int mi = 0; mi < 4; ++mi)
#pragma unroll
        for (int ni = 0; ni < 4; ++ni)
          acc[mi][ni] = __builtin_amdgcn_wmma_f32_16x16x4_f32(
              /*neg_a=*/false, afrag[mi],
              /*neg_b=*/false, bfrag[ni],
              /*c_mod=*/(short)0, acc[mi][ni],
              /*reuse_a=*/false, /*reuse_b=*/false);
    }

    if (t + 1 < NTILES) {
      storeB_lds(buf ^ 1);   // waits on its global loads via data dependence
      wait_async0();         // this wave's async A copies have landed
    }
    __syncthreads();         // waits DScnt; cross-wave visibility
  }

  // ---- store: C/D layout (ISA 7.12.2): VGPR v -> row = v + 8*laneHalf ----
#pragma unroll
  for (int mi = 0; mi < 4; ++mi) {
#pragma unroll
    for (int ni = 0; ni < 4; ++ni) {
      const int row_base = m0 + waveM * 64 + mi * 16 + lh * 8;
      const int col      = n0 + waveN * 64 + ni * 16 + ll;
#pragma unroll
      for (int v = 0; v < 8; ++v)
        Out[(size_t)(row_base + v) * N_TOTAL + col] = acc[mi][ni][v];
    }
  }
}

extern "C" void kernel_launch(void* const* d_in, const int* in_sizes, int n_in,
                              void* d_out, int out_size, void* d_ws, size_t ws_size,
                              hipStream_t stream) {
  const float* x     = (const float*)d_in[0];  // [B,S,DIN] f32
  const float* baseT = (const float*)d_in[1];  // [DIN,DOUT] f32
  const int*   mask  = (const int*)d_in[2];    // [DIN,DOUT] i32
  const float* coeff = (const float*)d_in[3];  // [1] f32
  float* out = (float*)d_out;                  // [B,S,DOUT] f32

  const int M = in_sizes[0] / K_TOTAL;         // B*S = 8192

  dim3 grid(N_TOTAL / BN, M / BM);             // (16, 32)
  binarydiff_wmma_gemm<<<grid, dim3(NTHREADS), 0, stream>>>(x, baseT, mask,
                                                            coeff, out, M);
}